// GAT_46574625358106
// MI455X (gfx1250) — compile-verified
//
#include <hip/hip_runtime.h>
#include <hip/hip_bf16.h>

#define F_IN   128
#define HEADS  8
#define CH     8
#define HOUT   (HEADS * CH)   // 64
#define NEG_SLOPE 0.2f
#define NEGINF_KEY 0x007FFFFFu  // fkey(-inf)

#if __has_builtin(__builtin_amdgcn_global_load_async_to_lds_b128) && \
    __has_builtin(__builtin_amdgcn_s_wait_asynccnt)
#define USE_ASYNC_LDS 1
#else
#define USE_ASYNC_LDS 0
#endif

typedef float v2f __attribute__((ext_vector_type(2)));
typedef float v8f __attribute__((ext_vector_type(8)));
typedef int   v4i __attribute__((ext_vector_type(4)));

#define GLOBAL_AS __attribute__((address_space(1)))
#define LDS_AS    __attribute__((address_space(3)))

// ---- order-preserving float <-> uint key (for atomicMax-based segment max) ----
__device__ __forceinline__ unsigned fkey(float f) {
    unsigned u = __float_as_uint(f);
    return (u & 0x80000000u) ? ~u : (u | 0x80000000u);
}
__device__ __forceinline__ float funkey(unsigned k) {
    unsigned u = (k & 0x80000000u) ? (k & 0x7FFFFFFFu) : ~k;
    return __uint_as_float(u);
}
__device__ __forceinline__ float lrelu(float v) {
    return v > 0.0f ? v : NEG_SLOPE * v;
}

// ------------------------------------------------------------------
// 0) init scratch (atomics require clean state every call)
// ------------------------------------------------------------------
__global__ void init_kernel(unsigned* __restrict__ key1, float* __restrict__ den1,
                            float* __restrict__ acc,  unsigned* __restrict__ key2,
                            float* __restrict__ den2, float* __restrict__ num2, int n) {
    int i = blockIdx.x * blockDim.x + threadIdx.x;
    if (i < n * HOUT) acc[i] = 0.0f;
    if (i < n * HEADS) { key1[i] = NEGINF_KEY; den1[i] = 0.0f; }
    if (i < n) { key2[i] = NEGINF_KEY; den2[i] = 0.0f; num2[i] = 0.0f; }
}

// ------------------------------------------------------------------
// 1) H = X @ W1 via V_WMMA_F32_16X16X4_F32 (wave32, one 16x16 tile/wave).
//    Block = 4 waves = the 4 column tiles of the 64-wide output.
//    The shared 16x128 A tile is staged once per block into LDS with
//    gfx1250 async global->LDS loads (ASYNCcnt), then read via ds_load.
// ------------------------------------------------------------------
__global__ __launch_bounds__(128) void gemm1_wmma(const float* __restrict__ X,
                                                  const float* __restrict__ W,
                                                  float* __restrict__ H, int n) {
    __shared__ float As[16 * F_IN];          // 8 KB

    const int tid  = threadIdx.x;
    const int wave = tid >> 5;               // 0..3 -> col tile
    const int lane = tid & 31;
    const int r    = lane & 15;
    const int half = lane >> 4;
    const int row0 = blockIdx.x * 16;
    const int col0 = wave * 16;

    // ---- cooperative A-tile stage: 16 rows x 128 cols = 512 float4 ----
#pragma unroll
    for (int stepi = 0; stepi < 4; ++stepi) {
        const int idx4 = stepi * 128 + tid;  // float4 index within tile
        const int arow = idx4 >> 5;          // 32 float4 per row
        const int acol = (idx4 & 31) * 4;
        int grow = row0 + arow;
        if (grow >= n) grow = n - 1;         // clamp (stores are guarded)
#if USE_ASYNC_LDS
        __builtin_amdgcn_global_load_async_to_lds_b128(
            (GLOBAL_AS v4i*)(X + (long)grow * F_IN + acol),
            (LDS_AS v4i*)(As + (size_t)idx4 * 4),
            0, 0);
#else
        float4 v = *(const float4*)(X + (long)grow * F_IN + acol);
        *(float4*)(As + (size_t)idx4 * 4) = v;
#endif
    }
#if USE_ASYNC_LDS
    __builtin_amdgcn_s_wait_asynccnt(0);
#endif
    __syncthreads();

    const float* __restrict__ arow = As + r * F_IN;

    v8f c = {};
#pragma unroll 4
    for (int k0 = 0; k0 < F_IN; k0 += 4) {
        const int ka = k0 + (half ? 2 : 0);   // ISA 7.12.2: lane-half selects K pair
        v2f a, b;
        a.x = arow[ka];                       // ds_load_b64
        a.y = arow[ka + 1];
        b.x = W[ka * HOUT + col0 + r];        // B: lane = column, VGPR/half = K row
        b.y = W[(ka + 1) * HOUT + col0 + r];
        c = __builtin_amdgcn_wmma_f32_16x16x4_f32(false, a, false, b,
                                                  (short)0, c, false, false);
    }

    // C: VGPR j -> rows j / j+8 ; fast path for full tiles (N % 16 == 0 case)
    if (row0 + 16 <= n) {
#pragma unroll
        for (int j = 0; j < 8; ++j)
            H[(long)(row0 + j + 8 * half) * HOUT + col0 + r] = c[j];
    } else {
#pragma unroll
        for (int j = 0; j < 8; ++j) {
            const int row = row0 + j + 8 * half;
            if (row < n) H[(long)row * HOUT + col0 + r] = c[j];
        }
    }
}

// ------------------------------------------------------------------
// 2) per-(node,head) attention logits
// ------------------------------------------------------------------
__global__ void alpha1_kernel(const float* __restrict__ H,
                              const float* __restrict__ asrc,
                              const float* __restrict__ adst,
                              float* __restrict__ as1, float* __restrict__ ad1, int n) {
    int i = blockIdx.x * blockDim.x + threadIdx.x;   // i = node*HEADS + head
    if (i >= n * HEADS) return;
    const int h = i & (HEADS - 1);
    const float4* __restrict__ hp = (const float4*)(H + (long)i * CH);
    const float4 h0 = hp[0], h1 = hp[1];
    const float4* sp = (const float4*)(asrc + h * CH);
    const float4* dp = (const float4*)(adst + h * CH);
    const float4 s0 = sp[0], s1 = sp[1], d0 = dp[0], d1 = dp[1];
    as1[i] = h0.x * s0.x + h0.y * s0.y + h0.z * s0.z + h0.w * s0.w +
             h1.x * s1.x + h1.y * s1.y + h1.z * s1.z + h1.w * s1.w;
    ad1[i] = h0.x * d0.x + h0.y * d0.y + h0.z * d0.z + h0.w * d0.w +
             h1.x * d1.x + h1.y * d1.y + h1.z * d1.z + h1.w * d1.w;
}

// ------------------------------------------------------------------
// 3) layer-1 edge passes (self-loops synthesized for e >= E)
// ------------------------------------------------------------------
__device__ __forceinline__ void edge_sd(const long long* __restrict__ ei,
                                        long e, int E, int& s, int& d) {
    if (e < E) { s = (int)ei[e]; d = (int)ei[(long)E + e]; }
    else       { s = d = (int)(e - E); }
}

__global__ __launch_bounds__(256) void edge_max1(const long long* __restrict__ ei,
                          int E, int etot,
                          const float* __restrict__ as1, const float* __restrict__ ad1,
                          unsigned* __restrict__ key1) {
    long i = (long)blockIdx.x * blockDim.x + threadIdx.x;   // (edge,head)
    if (i >= (long)etot * HEADS) return;
    const int h = (int)(i & (HEADS - 1));
    int s, d; edge_sd(ei, i >> 3, E, s, d);
    const float e = lrelu(as1[s * HEADS + h] + ad1[d * HEADS + h]);
    atomicMax(&key1[d * HEADS + h], fkey(e));
}

__global__ __launch_bounds__(256) void edge_acc1(const long long* __restrict__ ei,
                          int E, int etot,
                          const float* __restrict__ as1, const float* __restrict__ ad1,
                          const unsigned* __restrict__ key1, const float* __restrict__ H,
                          float* __restrict__ den1, float* __restrict__ acc) {
    long i = (long)blockIdx.x * blockDim.x + threadIdx.x;
    if (i >= (long)etot * HEADS) return;
    const int h = (int)(i & (HEADS - 1));
    int s, d; edge_sd(ei, i >> 3, E, s, d);
    const float e = lrelu(as1[s * HEADS + h] + ad1[d * HEADS + h]);
    const float w = expf(e - funkey(key1[d * HEADS + h]));
    atomicAdd(&den1[d * HEADS + h], w);
    const float4* __restrict__ hs = (const float4*)(H + ((long)s * HEADS + h) * CH);
    const float4 h0 = hs[0], h1 = hs[1];
    float* __restrict__ ap = acc + ((long)d * HEADS + h) * CH;
    atomicAdd(&ap[0], w * h0.x); atomicAdd(&ap[1], w * h0.y);
    atomicAdd(&ap[2], w * h0.z); atomicAdd(&ap[3], w * h0.w);
    atomicAdd(&ap[4], w * h1.x); atomicAdd(&ap[5], w * h1.y);
    atomicAdd(&ap[6], w * h1.z); atomicAdd(&ap[7], w * h1.w);
}

// ------------------------------------------------------------------
// 4) normalize + bias + ReLU + 64->1 projection + layer-2 logits, fused
// ------------------------------------------------------------------
__global__ void node1_kernel(const float* __restrict__ acc, const float* __restrict__ den1,
                             const float* __restrict__ b1,  const float* __restrict__ W2,
                             const float* __restrict__ asc2, const float* __restrict__ adc2,
                             float* __restrict__ t, float* __restrict__ as2,
                             float* __restrict__ ad2, int n) {
    int nn = blockIdx.x * blockDim.x + threadIdx.x;
    if (nn >= n) return;
    float proj = 0.0f;
#pragma unroll
    for (int h = 0; h < HEADS; ++h) {
        const float inv = 1.0f / (den1[nn * HEADS + h] + 1e-16f);
        const float* __restrict__ ap = acc + ((long)nn * HEADS + h) * CH;
#pragma unroll
        for (int cix = 0; cix < CH; ++cix) {
            float v = ap[cix] * inv + b1[h * CH + cix];
            v = v > 0.0f ? v : 0.0f;                  // ReLU (dropout = identity)
            proj += v * W2[h * CH + cix];
        }
    }
    t[nn]   = proj;
    as2[nn] = proj * asc2[0];
    ad2[nn] = proj * adc2[0];
}

// ------------------------------------------------------------------
// 5) layer-2 (single head) edge passes + final output
// ------------------------------------------------------------------
__global__ __launch_bounds__(256) void edge_max2(const long long* __restrict__ ei,
                          int E, int etot,
                          const float* __restrict__ as2, const float* __restrict__ ad2,
                          unsigned* __restrict__ key2) {
    long i = (long)blockIdx.x * blockDim.x + threadIdx.x;
    if (i >= etot) return;
    int s, d; edge_sd(ei, i, E, s, d);
    atomicMax(&key2[d], fkey(lrelu(as2[s] + ad2[d])));
}

__global__ __launch_bounds__(256) void edge_acc2(const long long* __restrict__ ei,
                          int E, int etot,
                          const float* __restrict__ as2, const float* __restrict__ ad2,
                          const unsigned* __restrict__ key2, const float* __restrict__ t,
                          float* __restrict__ den2, float* __restrict__ num2) {
    long i = (long)blockIdx.x * blockDim.x + threadIdx.x;
    if (i >= etot) return;
    int s, d; edge_sd(ei, i, E, s, d);
    const float e = lrelu(as2[s] + ad2[d]);
    const float w = expf(e - funkey(key2[d]));
    atomicAdd(&den2[d], w);
    atomicAdd(&num2[d], w * t[s]);
}

__global__ void final_kernel(const float* __restrict__ num2, const float* __restrict__ den2,
                             const float* __restrict__ b2, float* __restrict__ out, int n) {
    int nn = blockIdx.x * blockDim.x + threadIdx.x;
    if (nn >= n) return;
    out[nn] = num2[nn] / (den2[nn] + 1e-16f) + b2[0];
}

// ------------------------------------------------------------------
// launch
// ------------------------------------------------------------------
extern "C" void kernel_launch(void* const* d_in, const int* in_sizes, int n_in,
                              void* d_out, int out_size, void* d_ws, size_t ws_size,
                              hipStream_t stream) {
    const float*     x     = (const float*)d_in[0];
    const long long* ei    = (const long long*)d_in[1];   // int64 in reference
    const float*     W1    = (const float*)d_in[2];
    const float*     asrc1 = (const float*)d_in[3];
    const float*     adst1 = (const float*)d_in[4];
    const float*     b1    = (const float*)d_in[5];
    const float*     W2    = (const float*)d_in[6];
    const float*     asrc2 = (const float*)d_in[7];
    const float*     adst2 = (const float*)d_in[8];
    const float*     b2    = (const float*)d_in[9];
    float*           out   = (float*)d_out;

    const int N    = in_sizes[0] / F_IN;
    const int E    = in_sizes[1] / 2;
    const int etot = E + N;

    // ---- carve workspace (all offsets 256B-aligned) ----
    char* p = (char*)d_ws;
    auto carve = [&](size_t bytes) -> char* {
        char* r = p;
        p += (bytes + 255) & ~(size_t)255;
        return r;
    };
    float*    H    = (float*)   carve((size_t)N * HOUT  * 4);
    float*    as1  = (float*)   carve((size_t)N * HEADS * 4);
    float*    ad1  = (float*)   carve((size_t)N * HEADS * 4);
    unsigned* key1 = (unsigned*)carve((size_t)N * HEADS * 4);
    float*    den1 = (float*)   carve((size_t)N * HEADS * 4);
    float*    acc  = (float*)   carve((size_t)N * HOUT  * 4);
    float*    t    = (float*)   carve((size_t)N * 4);
    float*    as2  = (float*)   carve((size_t)N * 4);
    float*    ad2  = (float*)   carve((size_t)N * 4);
    unsigned* key2 = (unsigned*)carve((size_t)N * 4);
    float*    den2 = (float*)   carve((size_t)N * 4);
    float*    num2 = (float*)   carve((size_t)N * 4);
    (void)ws_size; (void)n_in; (void)out_size;

    const int B = 256;
    const long eh = (long)etot * HEADS;

    init_kernel<<<(N * HOUT + B - 1) / B, B, 0, stream>>>(key1, den1, acc, key2, den2, num2, N);
    gemm1_wmma<<<(N + 15) / 16, 128, 0, stream>>>(x, W1, H, N);
    alpha1_kernel<<<(N * HEADS + B - 1) / B, B, 0, stream>>>(H, asrc1, adst1, as1, ad1, N);
    edge_max1<<<(int)((eh + B - 1) / B), B, 0, stream>>>(ei, E, etot, as1, ad1, key1);
    edge_acc1<<<(int)((eh + B - 1) / B), B, 0, stream>>>(ei, E, etot, as1, ad1, key1, H, den1, acc);
    node1_kernel<<<(N + B - 1) / B, B, 0, stream>>>(acc, den1, b1, W2, asrc2, adst2, t, as2, ad2, N);
    edge_max2<<<(etot + B - 1) / B, B, 0, stream>>>(ei, E, etot, as2, ad2, key2);
    edge_acc2<<<(etot + B - 1) / B, B, 0, stream>>>(ei, E, etot, as2, ad2, key2, t, den2, num2);
    final_kernel<<<(N + B - 1) / B, B, 0, stream>>>(num2, den2, b2, out, N);
}